// FixedXMixing_77713138253958
// MI455X (gfx1250) — compile-verified
//
#include <hip/hip_runtime.h>

// FixedXMixing: T = (I + i*R)/sqrt(2) applied 22 times, R = reversal.
// T^2 = i*R, T^8 = I  =>  T^22 = T^6 = -i*R. So:
//   out_real[b,j] =  x_imag[b, dim-1-j]
//   out_imag[b,j] = -x_real[b, dim-1-j]
// Pure bandwidth problem: 256 MB in + 256 MB out -> ~22 us floor at 23.3 TB/s.
// Strategy: b128 non-temporal loads/stores (touch-once stream > 192 MB L2),
// 4 complex elements per lane, wave32-contiguous regions, per-wave
// global_prefetch_b8 ~64 KB ahead of the descending read stream.
// 2-D grid: blockIdx.y = batch row (kills per-thread 64-bit row math).

typedef __attribute__((ext_vector_type(4))) float v4f;

__global__ __launch_bounds__(256) void fixedx_mixing_kernel(
    const float* __restrict__ xr,   // [batch, dim] float32
    const float* __restrict__ xi,   // [batch, dim] float32
    float* __restrict__ out,        // [batch, dim] complex64 -> interleaved f32
    unsigned dim)                   // 2^22
{
    unsigned b  = blockIdx.y;                       // batch row (0..7)
    unsigned t  = blockIdx.x * 256u + threadIdx.x;  // thread within row
    unsigned j0 = t << 2;                           // output start (4 cplx)
    unsigned s0 = dim - 4u - j0;                    // reversed source start

    size_t rowOff = (size_t)(b * dim);              // < 2^25, scalar mul
    const float* srcR = xr + rowOff + s0;
    const float* srcI = xi + rowOff + s0;

    // One prefetch per wave, ~64 KB ahead in the (descending) read stream.
    if ((threadIdx.x & 31u) == 0u) {
        unsigned pf = (s0 >= 16384u) ? (s0 - 16384u) : 0u;
        __builtin_prefetch(xr + rowOff + pf, 0, 1);
        __builtin_prefetch(xi + rowOff + pf, 0, 1);
    }

    // 128-bit NT loads: source indices s0..s0+3 map (reversed) to j0+3..j0.
    v4f r  = __builtin_nontemporal_load((const v4f*)srcR);
    v4f im = __builtin_nontemporal_load((const v4f*)srcI);

    // Interleave as complex64: out[2j] = re, out[2j+1] = im.
    v4f o0, o1;
    o0.x = im.w;  o0.y = -r.w;   // j0
    o0.z = im.z;  o0.w = -r.z;   // j0+1
    o1.x = im.y;  o1.y = -r.y;   // j0+2
    o1.z = im.x;  o1.w = -r.x;   // j0+3

    float* dst = out + (rowOff + (size_t)j0) * 2u;  // interleaved offset
    __builtin_nontemporal_store(o0, (v4f*)dst);
    __builtin_nontemporal_store(o1, (v4f*)(dst + 4));
}

extern "C" void kernel_launch(void* const* d_in, const int* in_sizes, int n_in,
                              void* d_out, int out_size, void* d_ws, size_t ws_size,
                              hipStream_t stream) {
    const float* xr = (const float*)d_in[0];   // x_real [8, 2^22] f32
    const float* xi = (const float*)d_in[1];   // x_imag [8, 2^22] f32
    float* out = (float*)d_out;                // [8, 2^22] complex64 interleaved

    const unsigned BATCH = 8u;                 // matches reference constant
    unsigned total = (unsigned)in_sizes[0];    // BATCH * dim = 2^25
    unsigned dim   = total / BATCH;            // 2^22
    unsigned perRow = dim >> 2;                // threads per row (4 cplx/thread)

    dim3 block(256);
    dim3 grid(perRow / 256u, BATCH);           // 4096 x 8, exact cover (no tail)

    fixedx_mixing_kernel<<<grid, block, 0, stream>>>(xr, xi, out, dim);
}